// TMRPCEN10_42949673606
// MI455X (gfx1250) — compile-verified
//
#include <hip/hip_runtime.h>

typedef __attribute__((ext_vector_type(2))) float v2f;
typedef __attribute__((ext_vector_type(4))) float v4f;
typedef __attribute__((ext_vector_type(8))) float v8f;

#define BDIM 16
#define FDIM 128
#define TDIM 2048
#define KDIM 10
#define SEGLEN 256
#define NSEG (TDIM / SEGLEN) /* 8 */
#define TPAD 260
#define EPSV 1e-5f
#define LOG2E 1.4426950408889634f

// Raw hardware base-2 transcendentals: single v_exp_f32 / v_log_f32.
__device__ __forceinline__ float fast_exp2(float v) {
#if __has_builtin(__builtin_amdgcn_exp2f)
  return __builtin_amdgcn_exp2f(v);
#else
  return exp2f(v);
#endif
}
__device__ __forceinline__ float fast_log2(float v) {
#if __has_builtin(__builtin_amdgcn_logf)
  return __builtin_amdgcn_logf(v);
#else
  return log2f(v);
#endif
}

// ---------------------------------------------------------------------------
// Phase 1: per-(b,f,k) wave computes the smoother state entering each of the
// 8 segments of 256 timesteps. carry[0] = x[b,f,0] (virtual state at t=-1 so
// that t=0 uses the uniform recurrence state = (1-s)*prev + s*x[t]).
// ---------------------------------------------------------------------------
__global__ __launch_bounds__(256) void pcen_phase1(
    const float* __restrict__ x, const float* __restrict__ s_log,
    float* __restrict__ carry) {
  int wave = blockIdx.x * 8 + (threadIdx.x >> 5);
  int lane = threadIdx.x & 31;
  int k = wave % KDIM;
  int f = (wave / KDIM) % FDIM;
  int b = wave / (KDIM * FDIM);

  float s = fast_exp2(s_log[k] * LOG2E);
  float one_s = 1.0f - s;
  float d32 = one_s;
#pragma unroll
  for (int i = 0; i < 5; ++i) d32 *= d32;      // (1-s)^32
  float d256 = d32;
#pragma unroll
  for (int i = 0; i < 3; ++i) d256 *= d256;    // (1-s)^256
  float lg2 = fast_log2(one_s);
  float wl = s * fast_exp2(lg2 * (float)(31 - lane));  // s*(1-s)^(31-lane)

  const float* xr = x + (size_t)(b * FDIM + f) * TDIM;
  float c = xr[0];
  size_t cb = (size_t)((b * FDIM + f) * KDIM + k) * NSEG;

#pragma unroll 1
  for (int j = 0; j < NSEG; ++j) {
    const float* xs = xr + j * SEGLEN;
    // lane-parallel Horner: acc = sum_q (1-s)^{32*q} * x[32q+lane]
    float acc = xs[224 + lane];
#pragma unroll
    for (int q = 6; q >= 0; --q) acc = fmaf(acc, d32, xs[q * 32 + lane]);
    float v = wl * acc;
#pragma unroll
    for (int off = 16; off >= 1; off >>= 1) v += __shfl_xor(v, off);
    if (lane == 0) carry[cb + j] = c;   // carry-IN of segment j
    c = fmaf(d256, c, v);               // state at end of segment j
  }
}

// ---------------------------------------------------------------------------
// Phase 2: workgroup = (b, f-tile of 16, segment of 256 t). 10 waves, one per
// k. Each 16-timestep block is a 16x16x16 matmul done as 4x WMMA f32 16x16x4:
//   C[t', col] = sum_tau s*(1-s)^{t'-tau} * x[col, tau]  +  (1-s)^{t'+1}*carry
// A operand = x tile (M=col rows of f, K=tau), B operand = per-k coefficient
// matrix (K=tau, N=t'). C layout: lane = t' (both halves), VGPR v = col (v /
// v+8). Carry chains across blocks via one shfl per C register.
// All transcendentals are raw v_log_f32 / v_exp_f32 (base-2 formulation).
// ---------------------------------------------------------------------------
__global__ __launch_bounds__(320) void pcen_phase2(
    const float* __restrict__ x, const float* __restrict__ s_log,
    const float* __restrict__ alpha_log, const float* __restrict__ delta_log,
    const float* __restrict__ r_log, const float* __restrict__ carry,
    float* __restrict__ out) {
  __shared__ float xs[16 * TPAD];  // 16 rows x 256 t (+pad vs bank conflicts)

  int wid = threadIdx.x >> 5;
  int lane = threadIdx.x & 31;
  int b = blockIdx.x >> 6;
  int ft = (blockIdx.x >> 3) & 7;
  int seg = blockIdx.x & 7;
  int fbase = ft * 16;
  int t0 = seg * SEGLEN;
  int half = lane >> 4;      // lane half
  int lcol = lane & 15;      // = t' for B/C layout

  // ---- cooperative stage of x[b, fbase..fbase+15, t0..t0+255] into LDS ----
  if (wid < 8) {
#pragma unroll
    for (int rr = 0; rr < 2; ++rr) {
      int row = wid * 2 + rr;
      const float* src = x + (size_t)(b * FDIM + fbase + row) * TDIM + t0;
#pragma unroll
      for (int cc = 0; cc < 2; ++cc) {
        int idx = (cc * 32 + lane) * 4;
        v4f d = *(const v4f*)(src + idx);
        *(v4f*)(&xs[row * TPAD + idx]) = d;
      }
    }
  }
  __syncthreads();

  int k = wid;  // one wave per smoothing coefficient
  float s = fast_exp2(s_log[k] * LOG2E);
  float one_s = 1.0f - s;
  float lg2 = fast_log2(one_s);  // log2(1-s)

  // B coefficient matrices, 4 K-slices of 4. B[tau', t'] layout mirrors A:
  // vgpr vb holds row tau = 4*slice + vb + 2*half, col t' = lcol.
  v2f Bm[4];
#pragma unroll
  for (int sl = 0; sl < 4; ++sl) {
#pragma unroll
    for (int vb = 0; vb < 2; ++vb) {
      int row = 4 * sl + vb + 2 * half;
      Bm[sl][vb] =
          (lcol >= row) ? s * fast_exp2(lg2 * (float)(lcol - row)) : 0.0f;
    }
  }
  float dec = fast_exp2(lg2 * (float)(lcol + 1));  // (1-s)^(t'+1)

  // per-column (f) constants, one per C register
  float nalpha_v[8], delta_v[8], r_v[8], dr_v[8], carry_v[8];
  unsigned oofs[8];
  int lbase[8];
#pragma unroll
  for (int v = 0; v < 8; ++v) {
    int col = v + 8 * half;
    int fv = fbase + col;
    nalpha_v[v] = -fast_exp2(alpha_log[fv] * LOG2E);   // -alpha
    delta_v[v] = fast_exp2(delta_log[fv] * LOG2E);
    r_v[v] = fast_exp2(r_log[fv] * LOG2E);
    dr_v[v] = fast_exp2(r_v[v] * fast_log2(delta_v[v]));  // delta^r
    carry_v[v] = carry[(size_t)((b * FDIM + fv) * KDIM + k) * NSEG + seg];
    oofs[v] = (unsigned)(((b * KDIM + k) * FDIM + fv) * TDIM + t0 + lcol);
    lbase[v] = col * TPAD + lcol;
  }
  int abase = lcol * TPAD + 2 * half;  // A operand: row = lcol, tau offset

#pragma unroll 1
  for (int tb = 0; tb < SEGLEN; tb += 16) {
    // A operand slices: x[row=lcol, tb + 4*sl + {0,1}] (+2 for upper half)
    v2f A[4];
#pragma unroll
    for (int sl = 0; sl < 4; ++sl)
      A[sl] = *(const v2f*)(&xs[abase + tb + 4 * sl]);

    // init accumulator with carry term, then 4 K-slices of WMMA
    v8f C;
#pragma unroll
    for (int v = 0; v < 8; ++v) C[v] = carry_v[v] * dec;
#pragma unroll
    for (int sl = 0; sl < 4; ++sl)
      C = __builtin_amdgcn_wmma_f32_16x16x4_f32(false, A[sl], false, Bm[sl],
                                                (short)0, C, false, false);

    // next carry = state at t'=15: lane 15 (cols 0-7) / lane 31 (cols 8-15)
#pragma unroll
    for (int v = 0; v < 8; ++v) carry_v[v] = __shfl(C[v], 15 + (lane & 16));

    // pointwise PCEN + store (lane = t' -> contiguous 64B per half)
#pragma unroll
    for (int v = 0; v < 8; ++v) {
      float sm = C[v];
      float xv = xs[lbase[v] + tb];
      // (sm+eps)^(-alpha) = exp2(-alpha * log2(sm+eps))
      float t1 = fast_exp2(nalpha_v[v] * fast_log2(sm + EPSV));
      float y = fmaf(xv, t1, delta_v[v]);
      // y^r - delta^r = exp2(r * log2(y)) - dr
      float p = fast_exp2(r_v[v] * fast_log2(y)) - dr_v[v];
      out[oofs[v] + tb] = p;
    }
  }
}

// ---------------------------------------------------------------------------
extern "C" void kernel_launch(void* const* d_in, const int* in_sizes, int n_in,
                              void* d_out, int out_size, void* d_ws,
                              size_t ws_size, hipStream_t stream) {
  (void)in_sizes; (void)n_in; (void)out_size; (void)ws_size;
  const float* x = (const float*)d_in[0];
  const float* s_log = (const float*)d_in[1];
  const float* alpha_log = (const float*)d_in[2];
  const float* delta_log = (const float*)d_in[3];
  const float* r_log = (const float*)d_in[4];
  float* out = (float*)d_out;
  float* carry = (float*)d_ws;  // 16*128*10*8 floats = 640 KB

  // 16*128*10 waves / 8 waves per block
  pcen_phase1<<<2560, 256, 0, stream>>>(x, s_log, carry);
  // (b=16) x (f-tiles=8) x (segments=8) blocks, 10 waves each
  pcen_phase2<<<1024, 320, 0, stream>>>(x, s_log, alpha_log, delta_log, r_log,
                                        carry, out);
}